// MLP_71399536328822
// MI455X (gfx1250) — compile-verified
//
#include <hip/hip_runtime.h>

// MLP 10 -> 5 -> (5x5)*8 -> 1 over N=4,194,304 rows, f32.
// Memory floor: ~185 MB @ 23.3 TB/s ~= 8 us; ~510 FLOPs/row makes a pure
// VALU version borderline compute-bound, so the eight 5x5 layers run on the
// matrix pipe via V_WMMA_F32_16X16X4_F32 (A = weights, B = 16-row batch).
//
// Trick 1 (no swizzles in the layer chain): A rows f+8 duplicate rows f, so
// each D VGPR f holds h[f][n] in BOTH lane halves; the next layer's B operand
// is a single v_cndmask (lo half from D0/D1, hi half from D2/D3).
// Trick 2 (no bias adds): the 2nd K-chunk uses slot K=1 as a constant-1
// activation whose weight column is the bias vector, so D already includes
// bias and only ReLU remains per feature.
// Trick 3 (1-op ReLU): LLVM folds fmed3(x,0,+inf) back to fmaxnum and then
// inserts a canonicalize (v_max v,v,v) on WMMA outputs; inline asm
// "v_max_num_f32 d, 0, s" bypasses that (inputs are finite here).

typedef __attribute__((ext_vector_type(2))) float v2f;
typedef __attribute__((ext_vector_type(8))) float v8f;

// lanes 0-15 <-> lanes 16-31 (xor_mask=0x10, and_mask=0x1f)
#define SWZ_SWAPX16 0x401F

__device__ __forceinline__ float swz16(float v) {
    int i = __builtin_bit_cast(int, v);
    i = __builtin_amdgcn_ds_swizzle(i, SWZ_SWAPX16);
    return __builtin_bit_cast(float, i);
}

// ReLU in exactly one VOP2 instruction, no canonicalize.
__device__ __forceinline__ float relu1(float v) {
    float r;
    asm("v_max_num_f32 %0, 0, %1" : "=v"(r) : "v"(v));
    return r;
}

// One 5->5 Linear(+bias)+ReLU layer on a 16-column tile.
// Inputs b0,b1 = B chunk0 {K0/K2, K1/K3}; b4 = feature-4 row; one = 1.0f.
// A chunk1 carries {W[:,4], bias} in lanes 0-15 and zeros in lanes 16-31.
__device__ __forceinline__ void mid_layer(float h[5],
                                          float b0, float b1, float b4,
                                          float one,
                                          float a0x, float a0y,
                                          float a1x, float a1y) {
    v2f A0 = {a0x, a0y};
    v2f A1 = {a1x, a1y};
    v2f B0 = {b0, b1};
    v2f B1 = {b4, one};             // K=2,3 columns of A1 are zero
    v8f c = {0.f, 0.f, 0.f, 0.f, 0.f, 0.f, 0.f, 0.f};
    v8f d = __builtin_amdgcn_wmma_f32_16x16x4_f32(
        false, A0, false, B0, (short)0, c, false, false);
    d = __builtin_amdgcn_wmma_f32_16x16x4_f32(
        false, A1, false, B1, (short)0, d, false, false);
    h[0] = relu1(d[0]);
    h[1] = relu1(d[1]);
    h[2] = relu1(d[2]);
    h[3] = relu1(d[3]);
    h[4] = relu1(d[4]);
}

__global__ __launch_bounds__(256) void mlp_wmma_kernel(
    const float* __restrict__ x,     // [N,10]
    const float* __restrict__ W1,    // [5,10]
    const float* __restrict__ b1,    // [5]
    const float* __restrict__ Wmid,  // [8,5,5]
    const float* __restrict__ bmid,  // [8,5]
    const float* __restrict__ W10,   // [1,5]
    const float* __restrict__ b10,   // [1]
    float* __restrict__ out,         // [N]
    int nTiles)                      // N/32
{
    const int lane = threadIdx.x & 31;
    const bool hi  = lane >= 16;
    const int  m   = lane & 15;

    // ---- uniform weights for layers 1 and 10: scalar s_load ----
    float w1[5][10], bb1[5];
#pragma unroll
    for (int f = 0; f < 5; ++f) {
        bb1[f] = b1[f];
#pragma unroll
        for (int k = 0; k < 10; ++k) w1[f][k] = W1[f * 10 + k];
    }
    float w10[5];
#pragma unroll
    for (int f = 0; f < 5; ++f) w10[f] = W10[f];
    const float bb10 = b10[0];

    // ---- per-lane A operands (duplicated rows: M=f+8 == M=f) ----
    // 16x4 f32 A layout: lanes 0-15 = {K0,K1}, lanes 16-31 = {K2,K3}, row m.
    const int  r     = m & 7;           // duplicated row index
    const bool valid = (r < 5);
    const int  rc    = valid ? r : 0;   // keep loads in-bounds, select after
    const int  k0    = hi ? 2 : 0;
    float A0a[8], A0b[8], A1a[8], A1b[8];
#pragma unroll
    for (int l = 0; l < 8; ++l) {
        float wa = Wmid[l * 25 + rc * 5 + k0];
        float wb = Wmid[l * 25 + rc * 5 + k0 + 1];
        float w4 = Wmid[l * 25 + rc * 5 + 4];
        float bs = bmid[l * 5 + rc];
        A0a[l] = valid ? wa : 0.0f;
        A0b[l] = valid ? wb : 0.0f;
        A1a[l] = (valid && !hi) ? w4 : 0.0f;   // chunk1 K=0: feature-4 col
        A1b[l] = (valid && !hi) ? bs : 0.0f;   // chunk1 K=1: bias col
    }
    const float one = 1.0f;

    const int gid    = blockIdx.x * blockDim.x + threadIdx.x;
    const int wave   = gid >> 5;
    const int nWaves = (gridDim.x * blockDim.x) >> 5;

    for (int t = wave; t < nTiles; t += nWaves) {
        const size_t base = (size_t)t * 32;
        const float* xr = x + (base + (size_t)lane) * 10;

        // prefetch next grid-stride tile (speculative; OOB dropped)
        __builtin_prefetch(x + (base + (size_t)nWaves * 32 + lane) * 10, 0, 0);

        // ---- stream row: 5x global_load_b64, non-temporal ----
        float xv[10];
#pragma unroll
        for (int i = 0; i < 5; ++i) {
            v2f p = __builtin_nontemporal_load((const v2f*)(xr + 2 * i));
            xv[2 * i]     = p.x;
            xv[2 * i + 1] = p.y;
        }

        // ---- layer 1 (10->5), per-lane VALU with SGPR weights ----
        float g[5];
#pragma unroll
        for (int f = 0; f < 5; ++f) {
            float acc = bb1[f];
#pragma unroll
            for (int k = 0; k < 10; ++k) acc = fmaf(xv[k], w1[f][k], acc);
            g[f] = fmaxf(acc, 0.0f);   // FMA result is canonical: 1 v_max
        }

        // ---- per-lane g -> two 16-column WMMA tiles ----
        // Swizzles are UNCONDITIONAL (full EXEC); selects are pure cndmask.
        float s0 = swz16(g[0]);
        float s1 = swz16(g[1]);
        float s2 = swz16(g[2]);
        float s3 = swz16(g[3]);
        float s4 = swz16(g[4]);

        float h0[5], h1[5];
        // tile0: batch rows base+0..15
        mid_layer(h0, hi ? s2 : g[0], hi ? s3 : g[1], g[4], one,
                  A0a[0], A0b[0], A1a[0], A1b[0]);
        // tile1: batch rows base+16..31
        mid_layer(h1, hi ? g[2] : s0, hi ? g[3] : s1, s4, one,
                  A0a[0], A0b[0], A1a[0], A1b[0]);

        // ---- layers 3..9: duplicated D layout -> B is pure cndmask ----
#pragma unroll
        for (int l = 1; l < 8; ++l) {
            mid_layer(h0, hi ? h0[2] : h0[0], hi ? h0[3] : h0[1], h0[4], one,
                      A0a[l], A0b[l], A1a[l], A1b[l]);
            mid_layer(h1, hi ? h1[2] : h1[0], hi ? h1[3] : h1[1], h1[4], one,
                      A0a[l], A0b[l], A1a[l], A1b[l]);
        }

        // ---- layer 10 (5->1): valid in all lanes (duplicated halves) ----
        float o0 = bb10, o1 = bb10;
#pragma unroll
        for (int f = 0; f < 5; ++f) {
            o0 = fmaf(w10[f], h0[f], o0);
            o1 = fmaf(w10[f], h1[f], o1);
        }
        // lane n (lo) = tile0 col n; lane 16+n (hi) = tile1 col n (duplicate)
        float ov = hi ? o1 : o0;
        __builtin_nontemporal_store(ov, out + base + (size_t)lane);
    }
}

extern "C" void kernel_launch(void* const* d_in, const int* in_sizes, int n_in,
                              void* d_out, int out_size, void* d_ws, size_t ws_size,
                              hipStream_t stream) {
    (void)n_in; (void)d_ws; (void)ws_size;
    const float* x    = (const float*)d_in[0];
    const float* W1   = (const float*)d_in[1];
    const float* b1   = (const float*)d_in[2];
    const float* Wmid = (const float*)d_in[3];
    const float* bmid = (const float*)d_in[4];
    const float* W10  = (const float*)d_in[5];
    const float* b10  = (const float*)d_in[6];
    float* out = (float*)d_out;

    const int N = in_sizes[0] / 10;       // rows of x
    const int nTiles = N / 32;            // 32 rows per wave-tile

    const int threads = 256;              // 8 wave32 per block
    const int blocks  = 1024;             // 8192 waves; grid-stride over tiles
    mlp_wmma_kernel<<<blocks, threads, 0, stream>>>(
        x, W1, b1, Wmid, bmid, W10, b10, out, nTiles);
    (void)out_size;
}